// VectorQuantizer_EMA_Reset_52183852647085
// MI455X (gfx1250) — compile-verified
//
#include <hip/hip_runtime.h>
#include <cmath>

typedef float v2f __attribute__((ext_vector_type(2)));
typedef float v8f __attribute__((ext_vector_type(8)));

#define N_PTS 32768
#define DIM   256
#define KC    4096
#define DECAYF 0.99f
#define OMDF   0.01f   // 1 - DECAY
#define EPSF   1e-7f

// ---- workspace layout (float units) ----
#define WS_EMB   0                        // K*D embedding_sum
#define WS_COUNT (KC*DIM)                 // K cluster_count
#define WS_PE    (WS_COUNT + KC)          // K prob_ema
#define WS_MASK  (WS_PE + KC)             // K reset mask
#define WS_CNORM (WS_MASK + KC)           // K |c|^2
#define WS_CIDX  (WS_CNORM + KC)          // N (int) code indices
#define WS_DMIN  (WS_CIDX + N_PTS)        // N min squared distances
#define WS_SCAL  (WS_DMIN + N_PTS)        // 8 scalars: [0]=remained_sum [1]=num_reset

// ---- output layout (float units, concatenated return tuple) ----
#define OFF_XOUT    ((size_t)0)
#define OFF_ONEHOT  ((size_t)N_PTS*DIM)                  // 8,388,608
#define OFF_CLOSS   (OFF_ONEHOT + (size_t)N_PTS*KC)      // 142,606,336
#define OFF_PERP    (OFF_CLOSS + 1)
#define OFF_NEWCB   (OFF_PERP + 1)
#define OFF_NEWPROB (OFF_NEWCB + (size_t)KC*DIM)

#define BSTRIDE 260                       // padded LDS row stride (floats)

// -------------------- zero accumulators --------------------
__global__ void k_zero(float* __restrict__ ws, int n) {
  int i = blockIdx.x * blockDim.x + threadIdx.x;
  if (i < n) ws[i] = 0.0f;          // emb + count contiguous
  if (i < 8) ws[WS_SCAL + i] = 0.0f;
}

// -------------------- codebook row norms --------------------
__global__ __launch_bounds__(256) void k_cnorm(const float* __restrict__ cb,
                                               float* __restrict__ ws) {
  __shared__ float s[256];
  int k = blockIdx.x, t = threadIdx.x;
  float v = cb[(size_t)k * DIM + t];
  s[t] = v * v;
  __syncthreads();
  for (int w = 128; w > 0; w >>= 1) {
    if (t < w) s[t] += s[t + w];
    __syncthreads();
  }
  if (t == 0) ws[WS_CNORM + k] = s[0];
}

// -------------------- WMMA distance + argmin --------------------
// Block: 256 threads = 8 waves, 128 x-rows (16 per wave; A panel in 128
// VGPRs/lane). All waves sweep the same k-tile; the 16x256 B panel is
// double-buffered in LDS: prefetch tile kt+1 (global_load_b128) overlaps the
// 64-WMMA burst on tile kt, one barrier per tile. Codebook read once/block.
__global__ __launch_bounds__(256)
void k_argmin(const float* __restrict__ x, const float* __restrict__ cb,
              const float* __restrict__ cnorm_g, int* __restrict__ cidx,
              float* __restrict__ dmin, float* __restrict__ count) {
  __shared__ float btile[2][16 * BSTRIDE];   // ping-pong B panels
  __shared__ float cand_v[8 * 16 * 16];
  __shared__ int   cand_i[8 * 16 * 16];
  __shared__ float pnorm[8 * 16 * 2];
  __shared__ float xnorm[128];

  const int t    = threadIdx.x;
  const int wave = t >> 5;
  const int lane = t & 31;
  const int nidx = lane & 15;   // A row within tile / B-C column
  const int half = lane >> 4;   // K-pair selector
  const int r0   = blockIdx.x * 128;
  const int arow = r0 + wave * 16 + nidx;

  // staging coords: thread t owns B row srow, 16-float column chunk scol
  const int srow = t >> 4;
  const int scol = (t & 15) << 4;
  const float4* cbs = (const float4*)&cb[(size_t)srow * DIM + scol];

  // ---- preload this wave's A panel (16 rows x 256) into registers ----
  v2f afrag[64];
  {
    const float* xp = &x[(size_t)arow * DIM + half * 2];
    float pn = 0.0f;
#pragma unroll
    for (int j = 0; j < 64; ++j) {
      v2f a; a[0] = xp[4 * j]; a[1] = xp[4 * j + 1];
      afrag[j] = a;
      pn += a[0] * a[0] + a[1] * a[1];
    }
    pnorm[wave * 32 + nidx * 2 + half] = pn;
  }

  // ---- stage B tile 0 into buffer 0 ----
  {
    float4 p0 = cbs[0], p1 = cbs[1], p2 = cbs[2], p3 = cbs[3];
    float4* bs = (float4*)&btile[0][srow * BSTRIDE + scol];
    bs[0] = p0; bs[1] = p1; bs[2] = p2; bs[3] = p3;
  }
  __syncthreads();
  if (t < 128) xnorm[t] = pnorm[t * 2] + pnorm[t * 2 + 1];

  float minval[8];
  int   minidx[8];
#pragma unroll
  for (int v = 0; v < 8; ++v) { minval[v] = 3.4e38f; minidx[v] = 0; }

  for (int kt = 0; kt < 256; ++kt) {
    const int k0 = kt * 16;

    // prefetch next B tile into registers (overlaps WMMA burst below)
    float4 p0, p1, p2, p3;
    if (kt < 255) {
      const float4* nb = cbs + (size_t)(kt + 1) * (16 * DIM / 4);
      p0 = nb[0]; p1 = nb[1]; p2 = nb[2]; p3 = nb[3];
    }

    const float cn = cnorm_g[k0 + nidx];
    const float* bp = &btile[kt & 1][nidx * BSTRIDE + half * 2];
    v8f acc = {};
#pragma unroll
    for (int j = 0; j < 64; ++j) {
      v2f b; b[0] = bp[4 * j]; b[1] = bp[4 * j + 1];   // ds_load_2addr_b64
      acc = __builtin_amdgcn_wmma_f32_16x16x4_f32(false, afrag[j], false, b,
                                                  (short)0, acc, false, false);
    }
#pragma unroll
    for (int v = 0; v < 8; ++v) {
      float dd = cn - 2.0f * acc[v];       // |x|^2 added later (row-constant)
      if (dd < minval[v]) { minval[v] = dd; minidx[v] = k0 + nidx; }
    }

    if (kt < 255) {                        // write-side of ping-pong buffer
      float4* bs = (float4*)&btile[(kt + 1) & 1][srow * BSTRIDE + scol];
      bs[0] = p0; bs[1] = p1; bs[2] = p2; bs[3] = p3;
    }
    __syncthreads();                       // one barrier per k-tile
  }

#pragma unroll
  for (int v = 0; v < 8; ++v) {
    int rl = v + (half << 3);              // C layout: M = v + 8*(lane>=16)
    cand_v[wave * 256 + rl * 16 + nidx] = minval[v];
    cand_i[wave * 256 + rl * 16 + nidx] = minidx[v];
  }
  __syncthreads();

  if (t < 128) {                           // row t: reduce its 16 lane-candidates
    const int w = t >> 4, lr = t & 15;
    float bv = 3.4e38f; int bi = 0x7fffffff;
    for (int n2 = 0; n2 < 16; ++n2) {
      float v = cand_v[w * 256 + lr * 16 + n2];
      int  ii = cand_i[w * 256 + lr * 16 + n2];
      if (v < bv || (v == bv && ii < bi)) { bv = v; bi = ii; }
    }
    int n = r0 + t;
    cidx[n] = bi;
    dmin[n] = bv + xnorm[t];               // |x - c|^2
    atomicAdd(&count[bi], 1.0f);           // exact integer counts in f32
  }
}

// -------------------- embedding_sum scatter --------------------
__global__ void k_scatter(const float* __restrict__ x, const int* __restrict__ cidx,
                          float* __restrict__ ws) {
  int i = blockIdx.x * blockDim.x + threadIdx.x;   // i < N*D
  int n = i >> 8, d = i & 255;
  int idx = cidx[n];
  atomicAdd(&ws[WS_EMB + (size_t)idx * DIM + d], x[i]);
}

// -------------------- x_out (straight-through value), float4 --------------------
__global__ void k_xout(const float4* __restrict__ x4, const float4* __restrict__ cb4,
                       const int* __restrict__ cidx, float4* __restrict__ out4) {
  int i = blockIdx.x * blockDim.x + threadIdx.x;   // i < N*D/4
  int n = i >> 6, d4 = i & 63;
  float4 xv = x4[i];
  float4 q  = cb4[(size_t)cidx[n] * 64 + d4];
  float4 o;
  o.x = xv.x + (q.x - xv.x); o.y = xv.y + (q.y - xv.y);
  o.z = xv.z + (q.z - xv.z); o.w = xv.w + (q.w - xv.w);
  out4[i] = o;
}

// -------------------- one-hot codes, float4 stores --------------------
__global__ void k_onehot(const int* __restrict__ cidx, float4* __restrict__ out4) {
  size_t i = (size_t)blockIdx.x * blockDim.x + threadIdx.x;   // i < N*K/4
  int n = (int)(i >> 10);
  int k = (int)(i & 1023) << 2;
  int idx = cidx[n];
  float4 v;
  v.x = (k     == idx) ? 1.0f : 0.0f;
  v.y = (k + 1 == idx) ? 1.0f : 0.0f;
  v.z = (k + 2 == idx) ? 1.0f : 0.0f;
  v.w = (k + 3 == idx) ? 1.0f : 0.0f;
  out4[i] = v;
}

// -------------------- commitment loss (deterministic tree) --------------------
__global__ __launch_bounds__(256) void k_commit(const float* __restrict__ dmin,
                                                float* __restrict__ out) {
  __shared__ float s[256];
  int t = threadIdx.x;
  float acc = 0.0f;
  for (int i = t; i < N_PTS; i += 256) acc += dmin[i];
  s[t] = acc;
  __syncthreads();
  for (int w = 128; w > 0; w >>= 1) {
    if (t < w) s[t] += s[t + w];
    __syncthreads();
  }
  if (t == 0) out[OFF_CLOSS] = s[0] / ((float)N_PTS * (float)DIM);
}

// -------------------- perplexity, prob_ema, reset mask + sums --------------------
__global__ __launch_bounds__(256)
void k_stats(const float* __restrict__ count, const float* __restrict__ cprob,
             float* __restrict__ ws, float* __restrict__ out) {
  __shared__ float s_ent[256], s_rem[256], s_rst[256];
  int t = threadIdx.x;
  const float reset_prob = 0.01f / (float)KC;
  float ent = 0.0f, rem = 0.0f, rst = 0.0f;
  for (int k = t; k < KC; k += 256) {
    float c = count[k];
    float p = c / (float)N_PTS;                 // count.sum() == N exactly
    ent += p * logf(p + EPSF);
    float pe = cprob[k] * DECAYF + p * OMDF;
    float m  = (pe < reset_prob) ? 1.0f : 0.0f;
    ws[WS_PE + k]   = pe;
    ws[WS_MASK + k] = m;
    rem += (1.0f - m) * pe;
    rst += m;
  }
  s_ent[t] = ent; s_rem[t] = rem; s_rst[t] = rst;
  __syncthreads();
  for (int w = 128; w > 0; w >>= 1) {
    if (t < w) { s_ent[t] += s_ent[t + w]; s_rem[t] += s_rem[t + w]; s_rst[t] += s_rst[t + w]; }
    __syncthreads();
  }
  if (t == 0) {
    out[OFF_PERP]    = expf(-s_ent[0]);
    ws[WS_SCAL + 0]  = s_rem[0];
    ws[WS_SCAL + 1]  = s_rst[0];
  }
}

// -------------------- new cluster prob --------------------
__global__ void k_newprob(const float* __restrict__ ws, float* __restrict__ out) {
  int k = blockIdx.x * blockDim.x + threadIdx.x;
  if (k >= KC) return;
  float pe = ws[WS_PE + k], m = ws[WS_MASK + k];
  float rem_sum = ws[WS_SCAL + 0], nrst = ws[WS_SCAL + 1];
  float remained = (1.0f - m) * pe;
  out[OFF_NEWPROB + k] =
      m / (float)KC + remained / (rem_sum + EPSF) * (1.0f - nrst / (float)KC);
}

// -------------------- new codebook (EMA + dead-code reset) --------------------
__global__ void k_newcb(const float* __restrict__ ws, const float* __restrict__ cb,
                        const float* __restrict__ x, float* __restrict__ out) {
  int i = blockIdx.x * blockDim.x + threadIdx.x;   // i < K*D
  int k = i >> 8;
  float m   = ws[WS_MASK + k];
  float cnt = ws[WS_COUNT + k];
  float ema = cb[i] * DECAYF + (ws[WS_EMB + i] / (cnt + EPSF)) * OMDF;
  out[OFF_NEWCB + i] = (1.0f - m) * ema + m * x[i];  // x[:K] tile, i < K*D < N*D
}

extern "C" void kernel_launch(void* const* d_in, const int* in_sizes, int n_in,
                              void* d_out, int out_size, void* d_ws, size_t ws_size,
                              hipStream_t stream) {
  (void)in_sizes; (void)n_in; (void)out_size; (void)ws_size;
  const float* x     = (const float*)d_in[0];
  const float* cb    = (const float*)d_in[1];
  const float* cprob = (const float*)d_in[2];
  float* out = (float*)d_out;
  float* ws  = (float*)d_ws;
  int*   cidx = (int*)(ws + WS_CIDX);

  const int nzero = WS_COUNT + KC;                     // emb + count
  k_zero   <<<(nzero + 255) / 256, 256, 0, stream>>>(ws, nzero);
  k_cnorm  <<<KC, 256, 0, stream>>>(cb, ws);
  k_argmin <<<N_PTS / 128, 256, 0, stream>>>(x, cb, ws + WS_CNORM, cidx,
                                             ws + WS_DMIN, ws + WS_COUNT);
  k_scatter<<<(N_PTS * DIM) / 256, 256, 0, stream>>>(x, cidx, ws);
  k_xout   <<<(N_PTS * DIM / 4) / 256, 256, 0, stream>>>(
      (const float4*)x, (const float4*)cb, cidx, (float4*)(out + OFF_XOUT));
  k_onehot <<<(int)(((size_t)N_PTS * KC / 4) / 256), 256, 0, stream>>>(
      cidx, (float4*)(out + OFF_ONEHOT));
  k_commit <<<1, 256, 0, stream>>>(ws + WS_DMIN, out);
  k_stats  <<<1, 256, 0, stream>>>(ws + WS_COUNT, cprob, ws, out);
  k_newprob<<<KC / 256, 256, 0, stream>>>(ws, out);
  k_newcb  <<<(KC * DIM) / 256, 256, 0, stream>>>(ws, cb, x, out);
}